// VectorQuantize_17884243821134
// MI455X (gfx1250) — compile-verified
//
#include <hip/hip_runtime.h>
#include <stdint.h>

// Problem constants (from reference): x[N,C], codebook[K,C]
constexpr int N_ = 65536;
constexpr int C_ = 256;
constexpr int K_ = 4096;
constexpr int KT = K_ / 16;   // 256 codebook tiles of 16 rows

typedef __attribute__((ext_vector_type(2))) float v2f;
typedef __attribute__((ext_vector_type(8))) float v8f;

// ---------------------------------------------------------------------------
// Kernel 1: precompute squared norms  c2[k] = ||cb_k||^2 ,  x2[n] = ||x_n||^2
// ---------------------------------------------------------------------------
__global__ void vq_sqnorms(const float* __restrict__ x,
                           const float* __restrict__ cb,
                           float* __restrict__ c2,
                           float* __restrict__ x2) {
    int tid = blockIdx.x * blockDim.x + threadIdx.x;
    const float4* src;
    float* dst;
    int row;
    if (tid < K_) {
        row = tid;
        src = (const float4*)(cb + (size_t)row * C_);
        dst = c2 + row;
    } else if (tid < K_ + N_) {
        row = tid - K_;
        src = (const float4*)(x + (size_t)row * C_);
        dst = x2 + row;
    } else {
        return;
    }
    float s = 0.0f;
    #pragma unroll 8
    for (int j = 0; j < C_ / 4; ++j) {
        float4 v = src[j];
        s += v.x * v.x + v.y * v.y + v.z * v.z + v.w * v.w;
    }
    *dst = s;
}

// ---------------------------------------------------------------------------
// Kernel 2: WMMA argmin over  score(n,k) = c2[k] - 2 * <x_n, cb_k>
//   Block = 128 threads = 4 wave32; each wave owns a 16-row x tile.
//   Codebook tiles (16x256 = 16 KB) are double-buffered in LDS and streamed
//   with GLOBAL_LOAD_ASYNC_TO_LDS_B128 (ASYNCcnt), overlapping the
//   64-deep v_wmma_f32_16x16x4_f32 chain of the current tile.
// ---------------------------------------------------------------------------
__global__ __launch_bounds__(128, 1)
void vq_argmin(const float* __restrict__ x,
               const float* __restrict__ cb,
               const float* __restrict__ c2,
               const float* __restrict__ x2,
               float* __restrict__ outIdx,
               float* __restrict__ outMin) {
    __shared__ float cbT[2][16 * C_];   // 2 x 16 KB double-buffered cb tile

    const int lane = threadIdx.x & 31;
    const int wave = threadIdx.x >> 5;
    const int M    = lane & 15;      // A-matrix row / B-matrix column (N)
    const int hi   = lane >> 4;      // half-wave: K pair {0,1} vs {2,3}
    const int rowBase = (blockIdx.x * 4 + wave) * 16;

    // --- Async-stage one 16x256 codebook tile into LDS buffer `buf`.
    //     128 threads x 8 x b128 = 16 KB; tracked on ASYNCcnt.
    auto issue_tile = [&](int kt, int buf) {
        const float* src = cb + (size_t)kt * 16 * C_;
        #pragma unroll
        for (int j = 0; j < 8; ++j) {
            int e = (threadIdx.x + j * 128) * 4;            // float index
            uint32_t la = (uint32_t)(uintptr_t)&cbT[buf][e]; // LDS byte offset
            uint64_t ga = (uint64_t)(uintptr_t)(src + e);    // global address
            asm volatile("global_load_async_to_lds_b128 %0, %1, off"
                         :: "v"(la), "v"(ga) : "memory");
        }
    };

    // --- Load A fragments once: a[s] = x[rowBase+M][4s + 2*hi .. +1]
    // ISA 32-bit A 16x4 layout: lanes 0-15 -> K=0,1 ; lanes 16-31 -> K=2,3
    v2f a[64];
    {
        const float* xr = x + (size_t)(rowBase + M) * C_ + 2 * hi;
        #pragma unroll
        for (int s = 0; s < 64; ++s)
            a[s] = *(const v2f*)(xr + 4 * s);   // 8B-aligned -> global_load_b64
    }

    float best[8];
    int   bestK[8];
    #pragma unroll
    for (int v = 0; v < 8; ++v) { best[v] = 3.4e38f; bestK[v] = 0; }

    issue_tile(0, 0);                            // preload first tile

    for (int kt = 0; kt < KT; ++kt) {
        const int buf = kt & 1;

        // My async issues for tile kt are done; after the barrier everyone's
        // are, and all waves have finished reading buf^1 from iteration kt-1.
        asm volatile("s_wait_asynccnt 0x0" ::: "memory");
        __syncthreads();

        // Stream next tile into the other buffer while we compute this one.
        if (kt + 1 < KT) issue_tile(kt + 1, buf ^ 1);

        const float c2n = c2[kt * 16 + M];   // column-n norm for this lane

        // ---- 16x16 dot tile: 64 x v_wmma_f32_16x16x4_f32, K = 4 each
        v8f acc = {};
        #pragma unroll
        for (int s = 0; s < 64; ++s) {
            v2f b = *(const v2f*)&cbT[buf][M * C_ + 4 * s + 2 * hi]; // ds b64
            acc = __builtin_amdgcn_wmma_f32_16x16x4_f32(
                /*neg_a=*/false, a[s], /*neg_b=*/false, b,
                /*c_mod=*/(short)0, acc, /*reuse_a=*/false, /*reuse_b=*/false);
        }

        // ---- Running min/argmin.  D layout: VGPR v, lane L -> M=8*hi+v, N=M
        const int kcol = kt * 16 + M;
        #pragma unroll
        for (int v = 0; v < 8; ++v) {
            float sc = __builtin_fmaf(-2.0f, acc[v], c2n);
            if (sc < best[v]) { best[v] = sc; bestK[v] = kcol; }
        }
    }

    // ---- Cross-lane argmin over the 16 N-columns (xor masks stay in group)
    #pragma unroll
    for (int v = 0; v < 8; ++v) {
        float val = best[v];
        int   kk  = bestK[v];
        #pragma unroll
        for (int m = 8; m >= 1; m >>= 1) {
            float ov = __shfl_xor(val, m, 32);
            int   ok = __shfl_xor(kk,  m, 32);
            if (ov < val || (ov == val && ok < kk)) { val = ov; kk = ok; }
        }
        if ((lane & 15) == 0) {
            int row = rowBase + hi * 8 + v;
            outMin[row] = val + x2[row];     // add ||x||^2 back for min_dists
            outIdx[row] = (float)kk;         // k < 4096, exact in f32
        }
    }
}

// ---------------------------------------------------------------------------
// Kernel 3: gather  quantized[n, :] = codebook[idx[n], :]   (float4 copies)
// ---------------------------------------------------------------------------
__global__ void vq_gather(const float* __restrict__ cb,
                          const float* __restrict__ outIdx,
                          float* __restrict__ outQ) {
    int t  = blockIdx.x * blockDim.x + threadIdx.x;   // one float4 per thread
    int n  = t >> 6;                                  // C/4 = 64 float4 per row
    int c4 = t & 63;
    int k  = (int)outIdx[n];
    ((float4*)outQ)[t] = ((const float4*)cb)[k * 64 + c4];
}

// ---------------------------------------------------------------------------
extern "C" void kernel_launch(void* const* d_in, const int* in_sizes, int n_in,
                              void* d_out, int out_size, void* d_ws, size_t ws_size,
                              hipStream_t stream) {
    const float* x  = (const float*)d_in[0];   // [N, C]
    const float* cb = (const float*)d_in[1];   // [K, C]

    // d_out = quantized [N*C] ++ idx [N] ++ min_dists [N]  (return order)
    float* outQ   = (float*)d_out;
    float* outIdx = outQ + (size_t)N_ * C_;
    float* outMin = outIdx + N_;

    // workspace: c2 [K] ++ x2 [N]
    float* c2 = (float*)d_ws;
    float* x2 = c2 + K_;

    vq_sqnorms<<<(K_ + N_ + 255) / 256, 256, 0, stream>>>(x, cb, c2, x2);
    vq_argmin <<<N_ / 64, 128, 0, stream>>>(x, cb, c2, x2, outIdx, outMin);
    vq_gather <<<(N_ * (C_ / 4)) / 256, 256, 0, stream>>>(cb, outIdx, outQ);
}